// SubGraphLayer_5738076307802
// MI455X (gfx1250) — compile-verified
//
#include <hip/hip_runtime.h>

// ---------------------------------------------------------------------------
// Problem constants (from reference): B=16, N=50000, IN_C=128, HID=64, NC=512
// ---------------------------------------------------------------------------
#define Bn 16
#define Nn 50000
#define INC 128
#define HID 64
#define NC 512
#define ROWS (Bn * Nn)          // 800000
#define LN_EPS 1e-5f
#define WT_STRIDE 132           // padded K-stride (floats) for W^T in LDS -> conflict-free b64 reads
#define CHUNKS 16               // scatter-max chunks per batch

typedef float v2f __attribute__((ext_vector_type(2)));
typedef float v8f __attribute__((ext_vector_type(8)));

// ---------------------------------------------------------------------------
// K1: fused  h = relu(LN(x @ W + b))  -> out[:, :, 0:64]
// One workgroup = 8 waves; each wave owns a 16-row x 64-col tile.
// WMMA f32 16x16x4, K-loop of 32 steps; W^T staged in LDS.
// ---------------------------------------------------------------------------
__global__ __launch_bounds__(256) void k1_gemm_ln_relu(
    const float* __restrict__ x, const float* __restrict__ W,
    const float* __restrict__ bias, const float* __restrict__ gamma,
    const float* __restrict__ beta, float* __restrict__ out)
{
    __shared__ float Wt[HID * WT_STRIDE];      // 64*132*4 = 33792 B

    // Stage W transposed: Wt[col][k] = W[k][col]
    for (int i = threadIdx.x; i < INC * HID; i += 256) {
        int k = i >> 6;          // 0..127
        int c = i & 63;          // 0..63
        Wt[c * WT_STRIDE + k] = W[i];
    }
    __syncthreads();

    const int lane  = threadIdx.x & 31;
    const int wave  = threadIdx.x >> 5;        // 0..7
    const int l15   = lane & 15;
    const int phase = (lane < 16) ? 0 : 2;     // K sub-phase per ISA A-layout (16x4 f32)
    const int rowBase = blockIdx.x * 128 + wave * 16;   // exact: 6250*128 = 800000
    const int arow    = rowBase + l15;                  // row this lane feeds into A

    // Per-lane bias / gamma / beta for the 4 column tiles (col = nt*16 + l15)
    float bias_r[4], gam_r[4], bet_r[4];
#pragma unroll
    for (int nt = 0; nt < 4; ++nt) {
        bias_r[nt] = bias [nt * 16 + l15];
        gam_r[nt]  = gamma[nt * 16 + l15];
        bet_r[nt]  = beta [nt * 16 + l15];
    }

    v8f acc[4] = {};   // 4 tiles of 16x16 f32 accumulators

    const float* xrow = x + (size_t)arow * INC + phase;
    for (int k0 = 0; k0 < INC; k0 += 4) {
        // A fragment: x[arow, k0+phase], x[arow, k0+phase+1]   (8B aligned)
        v2f a = *(const v2f*)(xrow + k0);
#pragma unroll
        for (int nt = 0; nt < 4; ++nt) {
            // B fragment: W[k0+phase, col], W[k0+phase+1, col] via transposed LDS
            v2f bf = *(const v2f*)(&Wt[(nt * 16 + l15) * WT_STRIDE + k0 + phase]);
            acc[nt] = __builtin_amdgcn_wmma_f32_16x16x4_f32(
                false, a, false, bf, (short)0, acc[nt], false, false);
        }
    }

    // LayerNorm + ReLU. C/D layout: VGPR v -> row v (lanes 0-15) / row v+8 (lanes 16-31).
    const float inv64 = 1.0f / 64.0f;
#pragma unroll
    for (int v = 0; v < 8; ++v) {
        float vals[4];
        float s = 0.0f;
#pragma unroll
        for (int nt = 0; nt < 4; ++nt) {
            vals[nt] = acc[nt][v] + bias_r[nt];
            s += vals[nt];
        }
        // butterfly sum within each 16-lane half (rows are independent per half)
        s += __shfl_xor(s, 8, 32);
        s += __shfl_xor(s, 4, 32);
        s += __shfl_xor(s, 2, 32);
        s += __shfl_xor(s, 1, 32);
        float mu = s * inv64;

        float q = 0.0f;
#pragma unroll
        for (int nt = 0; nt < 4; ++nt) {
            vals[nt] -= mu;
            q += vals[nt] * vals[nt];
        }
        q += __shfl_xor(q, 8, 32);
        q += __shfl_xor(q, 4, 32);
        q += __shfl_xor(q, 2, 32);
        q += __shfl_xor(q, 1, 32);
        float inv = __frsqrt_rn(q * inv64 + LN_EPS);

        int r = rowBase + v + ((lane < 16) ? 0 : 8);
        size_t o = (size_t)r * (2 * HID);
#pragma unroll
        for (int nt = 0; nt < 4; ++nt) {
            float t = vals[nt] * inv * gam_r[nt] + bet_r[nt];
            t = (t > 0.0f) ? t : 0.0f;      // relu (never writes -0)
            out[o + nt * 16 + l15] = t;
        }
    }
}

// ---------------------------------------------------------------------------
// K2: per-batch scatter-max into seg[B][512][64].
// LDS-resident 512x64 table per workgroup (128 KB), merged with i32 atomic max
// (valid: h >= 0 post-ReLU, bit pattern is order-isomorphic; 0-init OK since
// every gathered cluster is non-empty).
// ---------------------------------------------------------------------------
__global__ __launch_bounds__(256) void k2_scatter_max(
    const float* __restrict__ out, const int* __restrict__ cluster,
    int* __restrict__ seg)
{
    __shared__ int lseg[NC * HID];            // 131072 B
    for (int i = threadIdx.x; i < NC * HID; i += 256) lseg[i] = 0;
    __syncthreads();

    const int b     = blockIdx.x / CHUNKS;
    const int chunk = blockIdx.x % CHUNKS;
    const int nPer  = Nn / CHUNKS;            // 3125
    const int r0    = chunk * nPer;
    const int sub   = threadIdx.x >> 6;       // 0..3 rows in flight
    const int c     = threadIdx.x & 63;       // column

    for (int i = sub; i < nPer; i += 4) {
        int r    = b * Nn + r0 + i;
        int cid  = cluster[r];
        float hv = out[(size_t)r * (2 * HID) + c];   // h lives in cols 0..63
        atomicMax(&lseg[cid * HID + c], __float_as_int(hv));   // ds_max_i32
    }
    __syncthreads();

    int* gseg = seg + (size_t)b * NC * HID;
    for (int i = threadIdx.x; i < NC * HID; i += 256) {
        int v = lseg[i];
        if (v > 0) atomicMax(&gseg[i], v);    // global_atomic_max_i32
    }
}

// ---------------------------------------------------------------------------
// K3: gather-broadcast  out[:, :, 64:128] = seg[b, cluster[b,n], :]
// One float4 per thread; seg is 2 MB -> L2 resident.
// ---------------------------------------------------------------------------
__global__ __launch_bounds__(256) void k3_gather(
    const float* __restrict__ seg, const int* __restrict__ cluster,
    float* __restrict__ out)
{
    size_t g      = (size_t)blockIdx.x * 256 + threadIdx.x;  // B*N*16 threads total
    size_t rowIdx = g >> 4;                                  // b*N + n
    int    c4     = (int)(g & 15) << 2;                      // col group 0,4,...,60
    int    b      = (int)(rowIdx / Nn);
    int    cid    = cluster[rowIdx];

    const float4 v = *(const float4*)(seg + ((size_t)b * NC + cid) * HID + c4);
    *(float4*)(out + rowIdx * (2 * HID) + HID + c4) = v;
}

// ---------------------------------------------------------------------------
extern "C" void kernel_launch(void* const* d_in, const int* in_sizes, int n_in,
                              void* d_out, int out_size, void* d_ws, size_t ws_size,
                              hipStream_t stream) {
    const float* x       = (const float*)d_in[0];
    const int*   cluster = (const int*)  d_in[1];
    const float* W       = (const float*)d_in[2];
    const float* bias    = (const float*)d_in[3];
    const float* gamma   = (const float*)d_in[4];
    const float* beta    = (const float*)d_in[5];
    float* out = (float*)d_out;
    int*   seg = (int*)d_ws;                  // [B][512][64] fp32-as-int, 2 MB

    // zero the cluster-max table every call (graph-capture safe)
    hipMemsetAsync(d_ws, 0, (size_t)Bn * NC * HID * sizeof(float), stream);

    k1_gemm_ln_relu<<<ROWS / 128, 256, 0, stream>>>(x, W, bias, gamma, beta, out);
    k2_scatter_max <<<Bn * CHUNKS, 256, 0, stream>>>(out, cluster, seg);
    k3_gather      <<<(ROWS * (HID / 4)) / 256, 256, 0, stream>>>((const float*)seg, cluster, out);
}